// RegressionLoss_33457795236470
// MI455X (gfx1250) — compile-verified
//
#include <hip/hip_runtime.h>

typedef __attribute__((ext_vector_type(2))) float v2f;
typedef __attribute__((ext_vector_type(8))) float v8f;

// Exact f32 sum of all 32 lane partials via V_WMMA_F32_16X16X4_F32.
// A (16x4 f32, 2 VGPRs/lane): lanes 0-15 -> A[m][0]=partial, A[m][1]=0;
// lanes 16-31 -> A[m][2]=partial, A[m][3]=0. B = all ones, C = 0.
// D[m][n] = partial[m] + partial[m+16]. Lane n (0-15) holds D[0..7][n] in its
// 8 D regs; lane 16+n holds D[8..15][n]. Per-lane sum of 8 regs + shfl_xor(16)
// gives the total on every lane. Multiplication by 1.0f is exact in f32.
__device__ __forceinline__ float wave_reduce_f32_wmma(float partial) {
    v2f a; a[0] = partial; a[1] = 0.0f;
    v2f b; b[0] = 1.0f;    b[1] = 1.0f;
    v8f c = {};
    v8f d = __builtin_amdgcn_wmma_f32_16x16x4_f32(
        /*neg_a=*/false, a, /*neg_b=*/false, b,
        /*c_mod=*/(short)0, c, /*reuse_a=*/false, /*reuse_b=*/false);
    float s = 0.0f;
#pragma unroll
    for (int i = 0; i < 8; ++i) s += d[i];
    s += __shfl_xor(s, 16, 32);   // combine rows 0-7 half with rows 8-15 half
    return s;
}

__global__ void rloss_init(float* __restrict__ acc) {
    if (threadIdx.x < 3) acc[threadIdx.x] = 0.0f;
}

__global__ void rloss_main(const float* __restrict__ pred,   // (B,Q,7)
                           const float* __restrict__ tgt,    // (N,7)
                           const int*   __restrict__ bidx,   // (N,)
                           const int*   __restrict__ sidx,   // (N,)
                           int N, int Q,
                           float* __restrict__ acc) {        // 3 f32 accumulators
    float pb = 0.0f, pg = 0.0f, pr = 0.0f;

    for (int i = blockIdx.x * blockDim.x + threadIdx.x; i < N;
         i += gridDim.x * blockDim.x) {
        const float* m = pred + ((size_t)bidx[i] * (size_t)Q + (size_t)sidx[i]) * 7;
        const float* t = tgt + (size_t)i * 7;

        float sm[3], sM[3], tm[3], tM[3];
        float l1 = 0.0f;
#pragma unroll
        for (int d = 0; d < 3; ++d) {
            float sc = m[d], sd = m[d + 3];
            float tc = t[d], td = t[d + 3];
            l1 += fabsf(sc - tc) + fabsf(sd - td);
            sm[d] = sc - 0.5f * sd;  sM[d] = sc + 0.5f * sd;
            tm[d] = tc - 0.5f * td;  tM[d] = tc + 0.5f * td;
        }

        float vol1 = 1.0f, vol2 = 1.0f, inter = 1.0f, volc = 1.0f;
#pragma unroll
        for (int d = 0; d < 3; ++d) {
            vol1 *= (sM[d] - sm[d]);
            vol2 *= (tM[d] - tm[d]);
            float lt = fmaxf(sm[d], tm[d]);
            float rb = fminf(sM[d], tM[d]);
            inter *= fmaxf(rb - lt, 0.0f);
            float ltc = fminf(sm[d], tm[d]);
            float rbc = fmaxf(sM[d], tM[d]);
            volc *= (rbc - ltc);
        }
        float uni  = vol1 + vol2 - inter;
        float iou  = inter / uni;
        float giou = iou - (volc - uni) / volc;

        pb += l1;
        pg += 1.0f - giou;
        pr += fabsf(m[6] - t[6]);
    }

    // Wave-level reduction via WMMA (EXEC all-1s here: executed unconditionally).
    float tb = wave_reduce_f32_wmma(pb);
    float tg = wave_reduce_f32_wmma(pg);
    float tr = wave_reduce_f32_wmma(pr);

    __shared__ float red[3][8];   // up to 8 waves per 256-thread block
    const int wave = threadIdx.x >> 5;
    const int lane = threadIdx.x & 31;
    if (lane == 0) { red[0][wave] = tb; red[1][wave] = tg; red[2][wave] = tr; }
    __syncthreads();

    if (threadIdx.x == 0) {
        const int nw = blockDim.x >> 5;
        float s0 = 0.0f, s1 = 0.0f, s2 = 0.0f;
        for (int w = 0; w < nw; ++w) {
            s0 += red[0][w]; s1 += red[1][w]; s2 += red[2][w];
        }
        atomicAdd(&acc[0], s0);
        atomicAdd(&acc[1], s1);
        atomicAdd(&acc[2], s2);
    }
}

__global__ void rloss_finalize(const float* __restrict__ acc,
                               const int*   __restrict__ nb_ptr,
                               float* __restrict__ out) {
    if (threadIdx.x == 0) {
        float nb = (float)nb_ptr[0];
        out[0] = acc[0] / nb;   // loss_bbox
        out[1] = acc[1] / nb;   // loss_giou
        out[2] = acc[2] / nb;   // loss_rad
    }
}

extern "C" void kernel_launch(void* const* d_in, const int* in_sizes, int n_in,
                              void* d_out, int out_size, void* d_ws, size_t ws_size,
                              hipStream_t stream) {
    const float* pred = (const float*)d_in[0];   // (16,1024,7) f32
    const float* tgt  = (const float*)d_in[1];   // (N,7) f32
    const int*   bid  = (const int*)  d_in[2];   // (N,) i32
    const int*   sid  = (const int*)  d_in[3];   // (N,) i32
    const int*   nbp  = (const int*)  d_in[4];   // scalar num_boxes

    const int N = in_sizes[1] / 7;               // target_boxes rows
    const int Q = 1024;                          // fixed by harness: pred_boxes (16,1024,7)

    float* acc = (float*)d_ws;                   // 3 f32 accumulators

    rloss_init<<<1, 32, 0, stream>>>(acc);

    const int threads = 256;
    int blocks = (N + threads - 1) / threads;
    if (blocks > 256) blocks = 256;
    if (blocks < 1)   blocks = 1;
    rloss_main<<<blocks, threads, 0, stream>>>(pred, tgt, bid, sid, N, Q, acc);

    rloss_finalize<<<1, 32, 0, stream>>>(acc, nbp, (float*)d_out);
}